// Learned2dRelativeSelfAttention_11003706212620
// MI455X (gfx1250) — compile-verified
//
#include <hip/hip_runtime.h>
#include <hip/hip_bf16.h>

typedef float v2f __attribute__((ext_vector_type(2)));
typedef float v8f __attribute__((ext_vector_type(8)));

static __device__ __forceinline__ v8f wmma4(v2f a, v2f b, v8f c) {
    // D = A(16x4,f32) * B(4x16,f32) + C(16x16,f32)
    return __builtin_amdgcn_wmma_f32_16x16x4_f32(
        /*neg_a=*/false, a, /*neg_b=*/false, b,
        /*c_mod=*/(short)0, c, /*reuse_a=*/false, /*reuse_b=*/false);
}

// Async copy of 16 bytes per lane: global -> LDS, tracked by ASYNCcnt.
// Operand order per CDNA5 ISA 15.18.3: VDST = LDS byte address, VADDR = 64-bit
// global address (GV mode, SADDR=off).
static __device__ __forceinline__ void async_copy_b128(unsigned lds_off,
                                                       const float* g) {
    asm volatile("global_load_async_to_lds_b128 %0, %1, off"
                 :
                 : "v"(lds_off), "v"((unsigned long long)(size_t)g)
                 : "memory");
}

#define WAIT_ASYNCCNT(n) asm volatile("s_wait_asynccnt %0" :: "n"(n) : "memory")

// ---------------------------------------------------------------------------
// Generic fp32 WMMA GEMM: C[M,N] = A[M,K] * B[K,N], row-major, M%16==0,
// N%16==0, K%4==0. One wave computes one 16x16 tile; 8 waves per block.
// ---------------------------------------------------------------------------
__global__ __launch_bounds__(256)
void gemm_wmma_f32(const float* __restrict__ A, const float* __restrict__ B,
                   float* __restrict__ C, int M, int N, int K) {
    int wave = (blockIdx.x * blockDim.x + threadIdx.x) >> 5;
    int lane = threadIdx.x & 31;
    int ntiles = N >> 4;
    int total  = (M >> 4) * ntiles;
    if (wave >= total) return;             // uniform per wave: EXEC stays all-1

    int rowt = wave / ntiles, colt = wave - rowt * ntiles;
    int row0 = rowt << 4, col0 = colt << 4;
    int half = lane >> 4;                  // K-pair selector
    int mn   = lane & 15;                  // A row (M) / B col (N)

    v8f acc = {0.f, 0.f, 0.f, 0.f, 0.f, 0.f, 0.f, 0.f};
    const float* arow = A + (size_t)(row0 + mn) * K;
    for (int kk = 0; kk < K; kk += 4) {
        int kb = kk + 2 * half;
        v2f a;  a.x  = arow[kb];                 a.y  = arow[kb + 1];
        v2f bb; bb.x = B[(size_t)kb * N + col0 + mn];
                bb.y = B[(size_t)(kb + 1) * N + col0 + mn];
        acc = wmma4(a, bb, acc);
    }
    // C/D layout: VGPR r -> row r (lanes 0-15) / row 8+r (lanes 16-31), col = lane%16
    for (int r = 0; r < 8; ++r)
        C[(size_t)(row0 + r + 8 * half) * N + col0 + mn] = acc[r];
}

// ---------------------------------------------------------------------------
// Relative-position bias:
//   rowb[b,x,y,h,u] = sum_d q[b,x,y,h,d]    * row_emb[min(x+u+31,62), d]
//   colb[b,x,y,h,v] = sum_d q[b,x,y,h,32+d] * col_emb[min(y+v+31,62), d]
// ---------------------------------------------------------------------------
__global__ __launch_bounds__(256)
void bias_kernel(const float* __restrict__ q, const float* __restrict__ row_emb,
                 const float* __restrict__ col_emb,
                 float* __restrict__ rowb, float* __restrict__ colb) {
    int tid = blockIdx.x * 256 + threadIdx.x;      // 2 * 2^20 threads
    int sel = tid >> 20;
    int r   = tid & 0xFFFFF;
    int u  = r & 31;
    int hh = (r >> 5) & 7;
    int y  = (r >> 8) & 31;
    int x  = (r >> 13) & 31;
    int b  = r >> 18;
    const float* qv = q + ((size_t)((b * 1024 + x * 32 + y) * 8 + hh)) * 64;
    float s = 0.f;
    if (sel == 0) {
        int e = x + u + 31; if (e > 62) e = 62;
        const float* em = row_emb + e * 32;
        for (int d = 0; d < 32; ++d) s += qv[d] * em[d];
        rowb[r] = s;
    } else {
        int e = y + u + 31; if (e > 62) e = 62;
        const float* em = col_emb + e * 32;
        for (int d = 0; d < 32; ++d) s += qv[32 + d] * em[d];
        colb[r] = s;
    }
}

// ---------------------------------------------------------------------------
// Stage one 16x64 f32 K tile and one 16x64 f32 V tile into LDS with async
// B128 copies (8 + 8 instructions; each lane moves 16B per instruction).
// ---------------------------------------------------------------------------
static __device__ __forceinline__ void stage_tile(const float* kbase,
                                                  const float* vbase,
                                                  float* sKbuf, float* sVbuf,
                                                  int lane) {
    for (int t = 0; t < 8; ++t) {
        int fo  = t * 128 + lane * 4;          // float offset within 16x64 tile
        int row = fo >> 6, col = fo & 63;
        async_copy_b128((unsigned)(size_t)(sKbuf + row * 64 + col),
                        kbase + (size_t)row * 512 + col);     // key stride 8*64
        async_copy_b128((unsigned)(size_t)(sVbuf + row * 64 + col),
                        vbase + (size_t)row * 2048 + col);    // v stride 32*64
    }
}

// ---------------------------------------------------------------------------
// Flash-style attention, 1 wave per (b, head, 16-query tile).
// Key index n = u*32 + v : logit = (q.k[b,u,v,h] + rowb[u] + colb[v]) / 8
// Value row for key n is hidden[b, v, u, :]  (spatial swap from reference).
// Output for query (x,y) is written at position (y,x)  (reference transpose).
// K/V tiles are double-buffered in LDS via async-to-LDS copies (ASYNCcnt).
// ---------------------------------------------------------------------------
__global__ __launch_bounds__(32)
void attn_kernel(const float* __restrict__ q, const float* __restrict__ k,
                 const float* __restrict__ hidden,
                 const float* __restrict__ rowb, const float* __restrict__ colb,
                 float* __restrict__ O) {
    __shared__ float sQ[16][64];
    __shared__ float sK[2][16][64];
    __shared__ float sV[2][16][64];
    __shared__ float sS[16][16];
    __shared__ float sP[16][16];
    __shared__ float sRB[16][32];
    __shared__ float sCB[16][32];
    __shared__ float sMax[16], sSum[16], sFac[16];

    int lane  = threadIdx.x;
    int qt    = blockIdx.x & 63;
    int hh    = (blockIdx.x >> 6) & 7;
    int b     = blockIdx.x >> 9;
    int qpos0 = qt << 4;
    int half  = lane >> 4;
    int mn    = lane & 15;

    // Load 16x64 Q tile (head hh)
    for (int t = 0; t < 32; ++t) {
        int idx = t * 32 + lane;
        int row = idx >> 6, col = idx & 63;
        sQ[row][col] = q[((size_t)((b * 1024 + qpos0 + row) * 8 + hh)) * 64 + col];
    }
    // Load per-query-row bias vectors
    for (int t = 0; t < 16; ++t) {
        int idx = t * 32 + lane;
        int row = idx >> 5, u = idx & 31;
        int qp = qpos0 + row;
        int x = qp >> 5, y = qp & 31;
        size_t base = (size_t)(((b * 32 + x) * 32 + y) * 8 + hh) * 32 + u;
        sRB[row][u] = rowb[base];
        sCB[row][u] = colb[base];
    }
    if (lane < 16) { sMax[lane] = -3.0e38f; sSum[lane] = 0.f; }
    __syncthreads();

    v8f acc[4];
    for (int g = 0; g < 4; ++g) acc[g] = (v8f){0.f,0.f,0.f,0.f,0.f,0.f,0.f,0.f};

    // Tile address helpers
    const size_t kstride_tile = (size_t)16 * 512;        // 16 keys * 512 floats
    const float* kb0 = k + ((size_t)((b * 1024) * 8 + hh)) * 64;

    // Prologue: stage tile 0 into buffer 0
    {
        int u = 0, v0 = 0;
        const float* vbase = hidden + ((size_t)((b * 32 + v0) * 32 + u)) * 64;
        stage_tile(kb0, vbase, &sK[0][0][0], &sV[0][0][0], lane);
    }

    for (int kt = 0; kt < 64; ++kt) {
        int buf = kt & 1;
        if (kt + 1 < 64) {
            // Prefetch next tile into the other buffer while we compute.
            int u = (kt + 1) >> 1, v0 = ((kt + 1) & 1) * 16;
            const float* vbase = hidden + ((size_t)((b * 32 + v0) * 32 + u)) * 64;
            stage_tile(kb0 + (size_t)(kt + 1) * kstride_tile, vbase,
                       &sK[buf ^ 1][0][0], &sV[buf ^ 1][0][0], lane);
            WAIT_ASYNCCNT(16);   // <=16 outstanding => tile kt has landed
        } else {
            WAIT_ASYNCCNT(0);
        }
        __syncthreads();

        const float* sKp = &sK[buf][0][0];
        const float* sVp = &sV[buf][0][0];

        // S = Q * K^T  (16 wmma steps over K=64), operands from LDS
        v8f s = {0.f,0.f,0.f,0.f,0.f,0.f,0.f,0.f};
        for (int kk = 0; kk < 64; kk += 4) {
            int kb = kk + 2 * half;
            v2f a;  a.x  = sQ[mn][kb];        a.y  = sQ[mn][kb + 1];
            v2f bb; bb.x = sKp[mn * 64 + kb]; bb.y = sKp[mn * 64 + kb + 1];
            s = wmma4(a, bb, s);
        }
        for (int r = 0; r < 8; ++r) sS[r + 8 * half][mn] = s[r];
        __syncthreads();

        // Online softmax update (one row per lane, lanes 0-15)
        if (lane < 16) {
            float tv[16];
            float mx = -3.0e38f;
            for (int n2 = 0; n2 < 16; ++n2) {
                int key = kt * 16 + n2;
                float val = (sS[lane][n2] + sRB[lane][key >> 5] + sCB[lane][key & 31]) * 0.125f;
                tv[n2] = val;
                mx = fmaxf(mx, val);
            }
            float om  = sMax[lane];
            float nm  = fmaxf(om, mx);
            float fac = __expf(om - nm);
            float sum = 0.f;
            for (int n2 = 0; n2 < 16; ++n2) {
                float p = __expf(tv[n2] - nm);
                sP[lane][n2] = p;
                sum += p;
            }
            sSum[lane] = sSum[lane] * fac + sum;
            sMax[lane] = nm;
            sFac[lane] = fac;
        }
        __syncthreads();

        // Rescale accumulators, then O += P * V  (4 groups x 4 wmma steps)
        float f[8];
        for (int r = 0; r < 8; ++r) f[r] = sFac[r + 8 * half];
        for (int g = 0; g < 4; ++g) {
            for (int r = 0; r < 8; ++r) acc[g][r] *= f[r];
            for (int kk = 0; kk < 4; ++kk) {
                int kb = kk * 4 + 2 * half;
                v2f a;  a.x  = sP[mn][kb];               a.y  = sP[mn][kb + 1];
                v2f bb; bb.x = sVp[kb * 64 + g * 16 + mn];
                        bb.y = sVp[(kb + 1) * 64 + g * 16 + mn];
                acc[g] = wmma4(a, bb, acc[g]);
            }
        }
        __syncthreads();
    }

    // Finalize: divide by row sums, store at transposed position (y,x)
    float inv[8];
    for (int r = 0; r < 8; ++r) inv[r] = 1.0f / sSum[r + 8 * half];
    for (int g = 0; g < 4; ++g) {
        for (int r = 0; r < 8; ++r) {
            int row = r + 8 * half;
            int qp  = qpos0 + row;
            int x = qp >> 5, y = qp & 31;
            O[((size_t)((b * 1024 + y * 32 + x) * 8 + hh)) * 64 + g * 16 + mn] =
                acc[g][r] * inv[r];
        }
    }
}

// ---------------------------------------------------------------------------
// Host entry
// ---------------------------------------------------------------------------
extern "C" void kernel_launch(void* const* d_in, const int* in_sizes, int n_in,
                              void* d_out, int out_size, void* d_ws, size_t ws_size,
                              hipStream_t stream) {
    const float* hidden  = (const float*)d_in[0];   // (4,32,32,64)
    const float* row_emb = (const float*)d_in[1];   // (63,32)
    const float* col_emb = (const float*)d_in[2];   // (63,32)
    const float* Wq      = (const float*)d_in[3];   // (64,512)
    const float* Wk      = (const float*)d_in[4];   // (64,512)
    const float* Wv      = (const float*)d_in[5];   // (512,64)
    float* out = (float*)d_out;                     // (4,32,32,64)

    float* ws   = (float*)d_ws;
    float* qb   = ws;                               // 4096*512
    float* kb   = qb + 4096 * 512;                  // 4096*512
    float* rwb  = kb + 4096 * 512;                  // 2^20
    float* clb  = rwb + (1 << 20);                  // 2^20
    float* Obuf = clb + (1 << 20);                  // 4096*512

    // Q/K projections: (4096x64)*(64x512); 8192 tiles / 8 waves -> 1024 blocks
    gemm_wmma_f32<<<1024, 256, 0, stream>>>(hidden, Wq, qb, 4096, 512, 64);
    gemm_wmma_f32<<<1024, 256, 0, stream>>>(hidden, Wk, kb, 4096, 512, 64);

    // Relative-position biases: 2*2^20 outputs
    bias_kernel<<<8192, 256, 0, stream>>>(qb, row_emb, col_emb, rwb, clb);

    // Attention: 4 batches * 8 heads * 64 query tiles = 2048 single-wave blocks
    attn_kernel<<<2048, 32, 0, stream>>>(qb, kb, hidden, rwb, clb, Obuf);

    // Output projection: (4096x512)*(512x64); 1024 tiles -> 128 blocks
    gemm_wmma_f32<<<128, 256, 0, stream>>>(Obuf, Wv, out, 4096, 64, 512);
}